// FeatureExtractor_2207613190279
// MI455X (gfx1250) — compile-verified
//
#include <hip/hip_runtime.h>
#include <stdint.h>

// FeatureExtractor: 3-channel searchsorted -> 1331-bin histogram over 16.7M
// pixels -> stable top-20 -> 20x4 float output.
//
// Memory-bound: 256 MB input stream @ 23.3 TB/s => ~11us floor. No matmul
// structure => no WMMA; the right CDNA5 paths are NT global_load_b128
// streaming, ds_add_u32 LDS-privatized histograms, global_atomic_add_u32
// merge, wave32 block reductions.

#define NBINS 11
#define HIST_SIZE (NBINS * NBINS * NBINS) // 1331
#define ITERS 20

typedef float v4f __attribute__((ext_vector_type(4)));

// ws layout (floats): [0..32] c01 (sorted/clipped cuts, 0..1 scale)
//                     [33..65] c255 (= c01 * 255, searchsorted boundaries)
//                     then uint32 hist[1331]

// ---------------------------------------------------------------------------
// Kernel 1: transform cuts (sort, clip, pin endpoints) + zero global hist.
// Rank-based sort of 11 elements per channel, tie-broken by original index
// (stable). clip is monotonic so clip(sort(c)) == sort(clip(c)) elementwise.
// ---------------------------------------------------------------------------
__global__ void __launch_bounds__(256)
prep_kernel(const float* __restrict__ r_cut, const float* __restrict__ g_cut,
            const float* __restrict__ b_cut, float* __restrict__ c01,
            float* __restrict__ c255, unsigned* __restrict__ hist) {
  __shared__ float s01[33];
  const int tid = threadIdx.x;

  if (blockIdx.x == 0) {
    if (tid < 33) {
      const int ch = tid / 11, i = tid % 11;
      const float* cut = (ch == 0) ? r_cut : (ch == 1) ? g_cut : b_cut;
      const float v = cut[i];
      int rank = 0;
      #pragma unroll
      for (int j = 0; j < 11; ++j) {
        const float w = cut[j];
        rank += (w < v) || (w == v && j < i); // stable rank
      }
      s01[ch * 11 + rank] = fminf(fmaxf(v, 0.0f), 1.0f);
    }
    __syncthreads();
    if (tid < 3) { // s[0]=0, s[-1]=1 per channel
      s01[tid * 11 + 0]  = 0.0f;
      s01[tid * 11 + 10] = 1.0f;
    }
    __syncthreads();
    if (tid < 33) {
      const float s = s01[tid];
      c01[tid]  = s;
      c255[tid] = s * 255.0f; // exactly the reference's rc * 255.0
    }
  }

  // All blocks: zero the global histogram (scratch is poisoned, and we must
  // re-zero every launch for determinism).
  for (int i = blockIdx.x * blockDim.x + tid; i < HIST_SIZE;
       i += gridDim.x * blockDim.x)
    hist[i] = 0u;
}

// ---------------------------------------------------------------------------
// Kernel 2: streaming histogram. NT b128 loads (stream > L2), searchsorted
// via 33 branchless compares against SGPR-resident cuts, ds_add_u32 into a
// per-block LDS histogram, skip-zero global_atomic_add_u32 flush.
// ---------------------------------------------------------------------------
__global__ void __launch_bounds__(256)
hist_kernel(const v4f* __restrict__ px, const float* __restrict__ c255,
            unsigned* __restrict__ hist, int n_pix) {
  __shared__ unsigned lh[HIST_SIZE];
  const int tid = threadIdx.x;

  for (int i = tid; i < HIST_SIZE; i += 256) lh[i] = 0u;

  // Wave-uniform cut values -> SGPRs; compares become v_cmp vs sgpr.
  float c[33];
  #pragma unroll
  for (int i = 0; i < 33; ++i) c[i] = c255[i];

  __syncthreads();

  const int stride = gridDim.x * 256;
  for (int i = blockIdx.x * 256 + tid; i < n_pix; i += stride) {
    const v4f p = __builtin_nontemporal_load(px + i); // b128, TH=NT

    // searchsorted(side='left') == count of boundaries strictly < x
    int riv = 0, giv = 0, biv = 0;
    #pragma unroll
    for (int j = 0; j < 11; ++j) {
      riv += (c[j]      < p.x);
      giv += (c[11 + j] < p.y);
      biv += (c[22 + j] < p.z);
    }
    // (iv - 1) mod 11 with Python semantics (iv in 0..10 here)
    const int rb = riv ? riv - 1 : 10;
    const int gb = giv ? giv - 1 : 10;
    const int bb = biv ? biv - 1 : 10;

    atomicAdd(&lh[rb * 121 + gb * 11 + bb], 1u); // ds_add_u32 (no rtn)
  }

  __syncthreads();
  for (int i = tid; i < HIST_SIZE; i += 256) {
    const unsigned v = lh[i];
    if (v) atomicAdd(&hist[i], v); // global_atomic_add_u32 (no rtn)
  }
}

// ---------------------------------------------------------------------------
// Kernel 3 (1 block): top-20 by (count desc, index asc) == stable
// argsort(-counts), then bin -> RGB centers. Keys pack count+1 in the high
// bits and (2047 - idx) low so smaller indices win ties; picked bins are set
// to 0xFFFFFFFF whose count field wraps to 0 and can never win again
// (1331 bins > 20 picks guarantees an unpicked candidate always exists).
// ---------------------------------------------------------------------------
__global__ void __launch_bounds__(256)
topk_kernel(const float* __restrict__ c01, const unsigned* __restrict__ hist,
            float* __restrict__ out) {
  __shared__ unsigned h[HIST_SIZE];
  __shared__ unsigned long long keys[256];
  __shared__ int winners[ITERS];
  const int tid = threadIdx.x;

  for (int i = tid; i < HIST_SIZE; i += 256) h[i] = hist[i];
  __syncthreads();

  for (int round = 0; round < ITERS; ++round) {
    unsigned long long best = 0ull;
    for (int b = tid; b < HIST_SIZE; b += 256) {
      const unsigned long long key =
          ((unsigned long long)(h[b] + 1u) << 16) |
          (unsigned long long)(unsigned)(2047 - b);
      best = (key > best) ? key : best;
    }
    keys[tid] = best;
    __syncthreads();
    #pragma unroll
    for (int s = 128; s > 0; s >>= 1) {
      if (tid < s) {
        const unsigned long long o = keys[tid + s];
        if (o > keys[tid]) keys[tid] = o;
      }
      __syncthreads();
    }
    if (tid == 0) {
      const int idx = 2047 - (int)(keys[0] & 0xFFFFull);
      winners[round] = idx;
      h[idx] = 0xFFFFFFFFu; // exclude: count field wraps to 0
    }
    __syncthreads();
  }

  if (tid < ITERS) {
    const int idx = winners[tid];
    int r = idx / 121;        // 0..10
    int g = (idx / 11) % 11;  // 0..10
    int b = idx % 11;         // 0..10
    r = (r > 9) ? 9 : r;      // clip to [0, n-2]
    g = (g > 9) ? 9 : g;
    b = (b > 9) ? 9 : b;
    // 255*r/11 + (rc[r+1]-rc[r])*255/2, association matching the reference
    // (x*255)/2 == x*255*0.5 bit-exactly (halving is exact).
    const float rv = 255.0f * (float)r / 11.0f +
                     (c01[r + 1] - c01[r]) * 255.0f * 0.5f;
    const float gv = 255.0f * (float)g / 11.0f +
                     (c01[11 + g + 1] - c01[11 + g]) * 255.0f * 0.5f;
    const float bv = 255.0f * (float)b / 11.0f +
                     (c01[22 + b + 1] - c01[22 + b]) * 255.0f * 0.5f;
    out[tid * 4 + 0] = rv;
    out[tid * 4 + 1] = gv;
    out[tid * 4 + 2] = bv;
    out[tid * 4 + 3] = 255.0f;
  }
}

extern "C" void kernel_launch(void* const* d_in, const int* in_sizes, int n_in,
                              void* d_out, int out_size, void* d_ws,
                              size_t ws_size, hipStream_t stream) {
  const v4f*  px    = (const v4f*)d_in[0];       // [N,4] f32
  const float* r_cut = (const float*)d_in[1];    // [11] f32
  const float* g_cut = (const float*)d_in[2];    // [11] f32
  const float* b_cut = (const float*)d_in[3];    // [11] f32
  const int n_pix = in_sizes[0] / 4;

  float*    c01  = (float*)d_ws;
  float*    c255 = c01 + 33;
  unsigned* hist = (unsigned*)(c255 + 33);

  prep_kernel<<<8, 256, 0, stream>>>(r_cut, g_cut, b_cut, c01, c255, hist);
  hist_kernel<<<2048, 256, 0, stream>>>(px, c255, hist, n_pix);
  topk_kernel<<<1, 256, 0, stream>>>(c01, hist, (float*)d_out);
}